// SVDLayer_38001870635878
// MI455X (gfx1250) — compile-verified
//
#include <hip/hip_runtime.h>

// CDNA5 / gfx1250, wave32. One workgroup (8 waves) per batch.
typedef __attribute__((ext_vector_type(2))) float v2f;
typedef __attribute__((ext_vector_type(4))) float f4;
typedef __attribute__((ext_vector_type(8))) float v8f;

#define NPTS  1024
#define TPB   256
#define WAVES 8            // TPB / 32
#define PPW   (NPTS/WAVES) // 128 points per wave

struct F2 { float s, p; }; // interleaved (src, pred) element

__global__ __launch_bounds__(TPB)
void svd_kabsch_kernel(const float* __restrict__ cors,
                       const float* __restrict__ src,
                       float* __restrict__ out)
{
    // interleaved (src,pred) per flat float index -> b64 gathers feed both A and B
    __shared__ __align__(16) F2   lds_sp[NPTS * 3];
    __shared__ float lds_red[WAVES * 16];
    __shared__ float lds_fin[16];

    const int b = blockIdx.x;
    const int t = threadIdx.x;

    // ---- Phase 1: stream batch tile (coalesced b128), fuse pred = src + cors ----
    {
        const f4* s4 = (const f4*)(src  + (size_t)b * NPTS * 3);
        const f4* c4 = (const f4*)(cors + (size_t)b * NPTS * 3);
        #pragma unroll
        for (int i = 0; i < 3; ++i) {
            int idx = t + i * TPB;          // 0..767 float4 chunks
            f4 sv = s4[idx];
            f4 cv = c4[idx];
            #pragma unroll
            for (int j = 0; j < 4; ++j) {
                F2 w; w.s = sv[j]; w.p = sv[j] + cv[j];
                lds_sp[idx * 4 + j] = w;    // ds_store_b64
            }
        }
    }
    __syncthreads();

    // ---- Phase 2: per-wave WMMA accumulation over its 128 points ----
    // A (16x4, M=dim rows + ones row at M=3), B (4x16, N=dim cols + ones col at N=3).
    // D[0:3][0:3]=Σ src⊗pred, D[r][3]=Σ src_r, D[3][c]=Σ pred_c, D[3][3]=count.
    const int lane = t & 31;
    const int half = lane >> 4;     // 0: K∈{0,1}; 1: K∈{2,3}
    const int m    = lane & 15;     // M for A rows / N for B cols
    const int msel = (m < 3) ? m : 0;
    const float fill = (m == 3) ? 1.0f : 0.0f;
    const int wave = t >> 5;
    const int base = wave * PPW;

    v8f acc0 = {};
    v8f acc1 = {};
    #pragma unroll 4
    for (int it = 0; it < PPW / 4; it += 2) {
        // --- chunk 0 ---
        int k0 = base + it * 4 + half * 2;
        F2 e0 = lds_sp[(k0    ) * 3 + msel];   // point K (this half), dim m
        F2 e1 = lds_sp[(k0 + 1) * 3 + msel];   // point K+1
        v2f A, B;
        A.x = (m < 3) ? e0.s : fill;  A.y = (m < 3) ? e1.s : fill;
        B.x = (m < 3) ? e0.p : fill;  B.y = (m < 3) ? e1.p : fill;
        acc0 = __builtin_amdgcn_wmma_f32_16x16x4_f32(false, A, false, B,
                                                     (short)0, acc0, false, false);
        // --- chunk 1 (independent accumulator for ILP) ---
        int k1 = k0 + 4;
        F2 f0 = lds_sp[(k1    ) * 3 + msel];
        F2 f1 = lds_sp[(k1 + 1) * 3 + msel];
        v2f A2, B2;
        A2.x = (m < 3) ? f0.s : fill;  A2.y = (m < 3) ? f1.s : fill;
        B2.x = (m < 3) ? f0.p : fill;  B2.y = (m < 3) ? f1.p : fill;
        acc1 = __builtin_amdgcn_wmma_f32_16x16x4_f32(false, A2, false, B2,
                                                     (short)0, acc1, false, false);
    }
    v8f acc = acc0 + acc1;   // same tile layout -> lanewise add

    // ---- Phase 3: stash D[0:4][0:4] per wave, reduce across waves ----
    if (lane < 4) {          // lanes 0-3 hold N=0-3; acc[v] holds row M=v (v<8)
        lds_red[wave * 16 + 0 * 4 + lane] = acc[0];
        lds_red[wave * 16 + 1 * 4 + lane] = acc[1];
        lds_red[wave * 16 + 2 * 4 + lane] = acc[2];
        lds_red[wave * 16 + 3 * 4 + lane] = acc[3];
    }
    __syncthreads();

    if (t < 16) {
        float s = 0.0f;
        #pragma unroll
        for (int w = 0; w < WAVES; ++w) s += lds_red[w * 16 + t];
        lds_fin[t] = s;
    }
    __syncthreads();

    // ---- Phase 4: one lane finalizes: h, polar-Newton, 4x4 transform ----
    if (t == 0) {
        float F[16];
        #pragma unroll
        for (int i = 0; i < 16; ++i) F[i] = lds_fin[i];
        const float invN = 1.0f / (float)NPTS;
        const float cs[3] = { F[3] * invN, F[7] * invN, F[11] * invN };
        const float cp[3] = { F[12] * invN, F[13] * invN, F[14] * invN };

        // centered cross-covariance: h = Σ s⊗p − (Σs)(Σp)ᵀ/N
        float X[3][3];
        #pragma unroll
        for (int r = 0; r < 3; ++r)
            #pragma unroll
            for (int c = 0; c < 3; ++c)
                X[r][c] = F[r * 4 + c] - F[r * 4 + 3] * F[12 + c] * invN;

        // Newton polar iteration with determinant scaling:
        // X <- 0.5*(mu*X + (1/mu)*X^{-T}),  mu = |det X|^{-1/3}
        // Converges to polar(h) = U Vᵀ; reference r_est = (U Vᵀ)ᵀ (no det fix).
        for (int iter = 0; iter < 8; ++iter) {
            float c00 = X[1][1]*X[2][2] - X[1][2]*X[2][1];
            float c01 = X[1][2]*X[2][0] - X[1][0]*X[2][2];
            float c02 = X[1][0]*X[2][1] - X[1][1]*X[2][0];
            float c10 = X[0][2]*X[2][1] - X[0][1]*X[2][2];
            float c11 = X[0][0]*X[2][2] - X[0][2]*X[2][0];
            float c12 = X[0][1]*X[2][0] - X[0][0]*X[2][1];
            float c20 = X[0][1]*X[1][2] - X[0][2]*X[1][1];
            float c21 = X[0][2]*X[1][0] - X[0][0]*X[1][2];
            float c22 = X[0][0]*X[1][1] - X[0][1]*X[1][0];
            float det = X[0][0]*c00 + X[0][1]*c01 + X[0][2]*c02;
            float ad  = fabsf(det);
            ad = ad < 1e-30f ? 1e-30f : ad;
            float sdet = (det < 0.0f) ? -ad : ad;
            float rdet = 1.0f / sdet;
            float mu   = exp2f(-0.3333333433f * log2f(ad));  // |det|^{-1/3}
            float hm   = 0.5f * mu;
            float rm   = 0.5f / mu;
            float Y[3][3] = {{c00,c01,c02},{c10,c11,c12},{c20,c21,c22}}; // cof(X)
            #pragma unroll
            for (int r = 0; r < 3; ++r)
                #pragma unroll
                for (int c = 0; c < 3; ++c)
                    X[r][c] = hm * X[r][c] + rm * (Y[r][c] * rdet); // X^{-T}=cof/det
        }

        // r_est = Xᵀ ;  t = c_pred − r_est · c_src ; emit 4x4 row-major
        float* o = out + (size_t)b * 16;
        #pragma unroll
        for (int i = 0; i < 3; ++i) {
            float r0 = X[0][i], r1 = X[1][i], r2 = X[2][i];
            o[i * 4 + 0] = r0;
            o[i * 4 + 1] = r1;
            o[i * 4 + 2] = r2;
            o[i * 4 + 3] = cp[i] - (r0 * cs[0] + r1 * cs[1] + r2 * cs[2]);
        }
        o[12] = 0.0f; o[13] = 0.0f; o[14] = 0.0f; o[15] = 1.0f;
    }
}

extern "C" void kernel_launch(void* const* d_in, const int* in_sizes, int n_in,
                              void* d_out, int out_size, void* d_ws, size_t ws_size,
                              hipStream_t stream) {
    (void)in_sizes; (void)n_in; (void)d_ws; (void)ws_size; (void)out_size;
    const float* cors = (const float*)d_in[0];  // [B, N, 3]
    const float* psrc = (const float*)d_in[1];  // [B, N, 3]
    float* out = (float*)d_out;                 // [B, 4, 4]
    const int B = 4096;
    svd_kabsch_kernel<<<B, TPB, 0, stream>>>(cors, psrc, out);
}